// SigmoidA_81681688035849
// MI455X (gfx1250) — compile-verified
//
#include <hip/hip_runtime.h>

typedef float v2f __attribute__((ext_vector_type(2)));
typedef float v8f __attribute__((ext_vector_type(8)));

constexpr int Bb  = 4;
constexpr int Ll  = 4096;
constexpr int Dd  = 2048;
constexpr int H   = 16;
constexpr int DH  = 64;
constexpr int KVD = 256;          // KVH*DH
constexpr int GD  = 1024;         // H*DH
constexpr int OC  = 3 * GD + H;   // 3088 output channels
constexpr int MV  = 2 * KVD + GD + H; // 1552 virtual weight rows (Wb|Wc|Wv|Wa)

constexpr int BM = 64;
constexpr int BN = 128;
constexpr int BK = 32;
constexpr int PITCH = 36;         // LDS row pitch in dwords (conflict-free b64 frags, 16B-aligned rows)
constexpr int KT = Dd / BK;       // 64 K-tiles

// ---- CDNA5 async global->LDS copy (ASYNCcnt-tracked, no VGPR staging) ----
__device__ __forceinline__ void async_b128(unsigned lds_off, const void* gptr) {
    asm volatile("global_load_async_to_lds_b128 %0, %1, off"
                 :: "v"(lds_off), "v"(gptr)
                 : "memory");
}
__device__ __forceinline__ void wait_async0() {
    asm volatile("s_wait_asynccnt 0x0" ::: "memory");
}

__device__ __forceinline__ const float* wRow(int vm,
                                             const float* __restrict__ Wa,
                                             const float* __restrict__ Wb,
                                             const float* __restrict__ Wc,
                                             const float* __restrict__ Wv) {
    if (vm >= MV) vm = MV - 1;            // clamp phantom rows; stores are masked
    if (vm < KVD)          return Wb + (size_t)vm * Dd;
    if (vm < 2 * KVD)      return Wc + (size_t)(vm - KVD) * Dd;
    if (vm < 2 * KVD + GD) return Wv + (size_t)(vm - 2 * KVD) * Dd;
    return Wa + (size_t)(vm - 2 * KVD - GD) * Dd;
}

__global__ __launch_bounds__(256)
void fused_proj_wmma(const float* __restrict__ x,
                     const float* __restrict__ Wa, const float* __restrict__ ba,
                     const float* __restrict__ Wb, const float* __restrict__ bb,
                     const float* __restrict__ Wc, const float* __restrict__ bc,
                     const float* __restrict__ Wv,
                     float* __restrict__ out) {
    __shared__ float sW[2][BM * PITCH];   // weight tile (M x K)
    __shared__ float sX[2][BN * PITCH];   // activation tile (N x K)

    const int tid  = threadIdx.x;
    const int lane = tid & 31;
    const int wave = tid >> 5;      // 0..7
    const int wm   = wave & 3;      // M sub-tile 0..3  (16 rows each)
    const int wn   = wave >> 2;     // N sub-block 0..1 (64 cols each)
    const int lo   = lane & 15;
    const int hi   = lane >> 4;

    const int n_base = blockIdx.x * BN;   // 0..4095
    const int m_base = blockIdx.y * BM;   // virtual row base
    const int bz     = blockIdx.z;        // batch

    const float* xb = x + (size_t)bz * Ll * Dd;

    // Issue one K-tile of async global->LDS copies (6 x b128 per thread).
    auto issueTile = [&](int k0, int buf) {
#pragma unroll
        for (int i = 0; i < 2; ++i) {
            int ch  = tid + i * 256;       // 0..511 : 64 rows x 8 float4
            int row = ch >> 3;
            int c4  = (ch & 7) << 2;
            const float* p = wRow(m_base + row, Wa, Wb, Wc, Wv) + k0 + c4;
            async_b128((unsigned)(uintptr_t)&sW[buf][row * PITCH + c4], p);
        }
#pragma unroll
        for (int i = 0; i < 4; ++i) {
            int ch  = tid + i * 256;       // 0..1023 : 128 rows x 8 float4
            int row = ch >> 3;
            int c4  = (ch & 7) << 2;
            const float* p = xb + (size_t)(n_base + row) * Dd + k0 + c4;
            async_b128((unsigned)(uintptr_t)&sX[buf][row * PITCH + c4], p);
        }
    };

    v8f acc[4];
#pragma unroll
    for (int j = 0; j < 4; ++j) acc[j] = v8f{0.f, 0.f, 0.f, 0.f, 0.f, 0.f, 0.f, 0.f};

    // prologue: stage tile 0
    issueTile(0, 0);
    wait_async0();
    __syncthreads();

    for (int kt = 0; kt < KT; ++kt) {
        const int cur   = kt & 1;
        const bool more = (kt + 1) < KT;
        if (more) issueTile((kt + 1) * BK, cur ^ 1);   // async into the other buffer

        // fragment base pointers in LDS (ISA 16x4 f32 A layout: lane.lo = M, half/vgpr = K)
        const float* pw = &sW[cur][(wm * 16 + lo) * PITCH + 2 * hi];
        const float* px = &sX[cur][(wn * 64 + lo) * PITCH + 2 * hi];

#pragma unroll
        for (int kk = 0; kk < BK / 4; ++kk) {
            v2f a = *(const v2f*)(pw + kk * 4);
#pragma unroll
            for (int j = 0; j < 4; ++j) {
                v2f b = *(const v2f*)(px + j * 16 * PITCH + kk * 4);
                acc[j] = __builtin_amdgcn_wmma_f32_16x16x4_f32(
                    /*neg_a=*/false, a, /*neg_b=*/false, b,
                    /*c_mod=*/(short)0, acc[j],
                    /*reuse_a=*/false, /*reuse_b=*/false);
            }
        }

        if (more) {
            wait_async0();     // this wave's async copies for buf^1 landed in LDS
            __syncthreads();   // whole workgroup's copies visible
        }
    }

    // Epilogue: route virtual rows -> output channels, add bias, 4x KV repeat for B/C.
    float* outb = out + (size_t)bz * OC * Ll;
    const int lane_n = n_base + wn * 64 + lo;

#pragma unroll
    for (int r = 0; r < 8; ++r) {
        const int vm = m_base + wm * 16 + 8 * hi + r;  // C/D layout: hi half holds M+8
        if (vm >= MV) continue;

        int c0, ncopy;
        float bias;
        if (vm < KVD) {                       // Bk -> B_gate [0, 1024)
            c0 = (vm >> 6) * 256 + (vm & 63); ncopy = 4; bias = bb[vm];
        } else if (vm < 2 * KVD) {            // Ck -> C [2048, 3072)
            const int vc = vm - KVD;
            c0 = 2 * GD + (vc >> 6) * 256 + (vc & 63); ncopy = 4; bias = bc[vc];
        } else if (vm < 2 * KVD + GD) {       // V [1024, 2048)
            c0 = GD + (vm - 2 * KVD); ncopy = 1; bias = 0.f;
        } else {                              // A [3072, 3088)
            const int va = vm - (2 * KVD + GD);
            c0 = 3 * GD + va; ncopy = 1; bias = ba[va];
        }

#pragma unroll
        for (int j = 0; j < 4; ++j) {
            const float val = acc[j][r] + bias;
            const size_t off = (size_t)c0 * Ll + (size_t)(lane_n + j * 16);
            for (int cp = 0; cp < ncopy; ++cp) {
                outb[off + (size_t)cp * DH * Ll] = val;  // repeat stride = 64 channels
            }
        }
    }
}

extern "C" void kernel_launch(void* const* d_in, const int* in_sizes, int n_in,
                              void* d_out, int out_size, void* d_ws, size_t ws_size,
                              hipStream_t stream) {
    const float* x  = (const float*)d_in[0];
    const float* Wa = (const float*)d_in[1];
    const float* ba = (const float*)d_in[2];
    const float* Wb = (const float*)d_in[3];
    const float* bb = (const float*)d_in[4];
    const float* Wc = (const float*)d_in[5];
    const float* bc = (const float*)d_in[6];
    const float* Wv = (const float*)d_in[7];
    float* out = (float*)d_out;

    dim3 grid(Ll / BN, (MV + BM - 1) / BM, Bb);  // (32, 25, 4)
    fused_proj_wmma<<<grid, 256, 0, stream>>>(x, Wa, ba, Wb, bb, Wc, bc, Wv, out);
}